// Paper3Loss_14001593385262
// MI455X (gfx1250) — compile-verified
//
#include <hip/hip_runtime.h>
#include <hip/hip_bf16.h>

typedef float v2f __attribute__((ext_vector_type(2)));
typedef float v8f __attribute__((ext_vector_type(8)));

#define EPSF 1e-6f
#define INF_BITS 0x7F800000u

#if __has_builtin(__builtin_amdgcn_global_load_async_to_lds_b32)
#define ASYNC_STAGE 1
#else
#define ASYNC_STAGE 0
#endif

// ---------------------------------------------------------------------------
// init: single write per word (zeros, +inf for colmin, all-ones for rowpacked)
// ---------------------------------------------------------------------------
__global__ void k_init(unsigned* ws, int nwords, int rp_off, int rp_len,
                       int col_off, int col_len) {
    int i = blockIdx.x * blockDim.x + threadIdx.x;
    if (i >= nwords) return;
    unsigned v = 0u;
    if (i >= rp_off && i < rp_off + rp_len) v = 0xFFFFFFFFu;
    else if (i >= col_off && i < col_off + col_len) v = INF_BITS;
    ws[i] = v;
}

// ---------------------------------------------------------------------------
// face normals scatter:  vn[f[k]] += cross(v1-v0, v2-v0)
// ---------------------------------------------------------------------------
__global__ void k_face_normals(const int* __restrict__ f, const float* __restrict__ v,
                               float* __restrict__ vn, int nf) {
    int i = blockIdx.x * blockDim.x + threadIdx.x;
    if (i >= nf) return;
    int i0 = f[3*i], i1 = f[3*i+1], i2 = f[3*i+2];
    float v0x = v[3*i0], v0y = v[3*i0+1], v0z = v[3*i0+2];
    float e1x = v[3*i1]   - v0x, e1y = v[3*i1+1] - v0y, e1z = v[3*i1+2] - v0z;
    float e2x = v[3*i2]   - v0x, e2y = v[3*i2+1] - v0y, e2z = v[3*i2+2] - v0z;
    float nx = e1y*e2z - e1z*e2y;
    float ny = e1z*e2x - e1x*e2z;
    float nz = e1x*e2y - e1y*e2x;
    atomicAdd(&vn[3*i0+0], nx); atomicAdd(&vn[3*i0+1], ny); atomicAdd(&vn[3*i0+2], nz);
    atomicAdd(&vn[3*i1+0], nx); atomicAdd(&vn[3*i1+1], ny); atomicAdd(&vn[3*i1+2], nz);
    atomicAdd(&vn[3*i2+0], nx); atomicAdd(&vn[3*i2+1], ny); atomicAdd(&vn[3*i2+2], nz);
}

// ---------------------------------------------------------------------------
// uniform-laplacian scatter
// ---------------------------------------------------------------------------
__global__ void k_lap_scatter(const int* __restrict__ f, const float* __restrict__ v,
                              float* __restrict__ nsum, float* __restrict__ deg, int nf) {
    int i = blockIdx.x * blockDim.x + threadIdx.x;
    if (i >= nf) return;
    int i0 = f[3*i], i1 = f[3*i+1], i2 = f[3*i+2];
    float x0 = v[3*i0], y0 = v[3*i0+1], z0 = v[3*i0+2];
    float x1 = v[3*i1], y1 = v[3*i1+1], z1 = v[3*i1+2];
    float x2 = v[3*i2], y2 = v[3*i2+1], z2 = v[3*i2+2];
    atomicAdd(&nsum[3*i0+0], x1 + x2); atomicAdd(&nsum[3*i0+1], y1 + y2); atomicAdd(&nsum[3*i0+2], z1 + z2);
    atomicAdd(&nsum[3*i1+0], x2 + x0); atomicAdd(&nsum[3*i1+1], y2 + y0); atomicAdd(&nsum[3*i1+2], z2 + z0);
    atomicAdd(&nsum[3*i2+0], x0 + x1); atomicAdd(&nsum[3*i2+1], y0 + y1); atomicAdd(&nsum[3*i2+2], z0 + z1);
    atomicAdd(&deg[i0], 2.0f); atomicAdd(&deg[i1], 2.0f); atomicAdd(&deg[i2], 2.0f);
}

// ---------------------------------------------------------------------------
// nearest-neighbour label sampling + positive count
// ---------------------------------------------------------------------------
__global__ void k_targets(const float* __restrict__ pv, const int* __restrict__ label,
                          float* __restrict__ targ, float* __restrict__ pc,
                          int np, int D, int H, int W) {
    int i = blockIdx.x * blockDim.x + threadIdx.x;
    if (i >= np) return;
    float x = pv[3*i], y = pv[3*i+1], z = pv[3*i+2];
    int ix = (int)rintf(x * (float)(W - 1));   // round-half-even == jnp.round
    int iy = (int)rintf(y * (float)(H - 1));
    int iz = (int)rintf(z * (float)(D - 1));
    bool inb = (ix >= 0) && (ix < W) && (iy >= 0) && (iy < H) && (iz >= 0) && (iz < D);
    int ixc = min(max(ix, 0), W - 1);
    int iyc = min(max(iy, 0), H - 1);
    int izc = min(max(iz, 0), D - 1);
    float t = 0.0f;
    if (inb) t = (label[(izc * H + iyc) * W + ixc] == 1) ? 1.0f : 0.0f;
    targ[i] = t;
    if (t > 0.5f) atomicAdd(pc, 1.0f);
}

// ---------------------------------------------------------------------------
// async (or fallback) staging of one 256-vertex chunk into LDS.
// Dwords are scattered into a float4-strided layout (w unused) so each tile
// later reads one aligned ds_load_b128.
// ---------------------------------------------------------------------------
__device__ __forceinline__ void stage_chunk(const float* __restrict__ gv, int base,
                                            int cnt, int cnt16, float4* sbuf, int tid) {
    const int ndw = 3 * cnt;
#if ASYNC_STAGE
    typedef __attribute__((address_space(1))) int* gint_t;   // global src
    typedef __attribute__((address_space(3))) int* lint_t;   // LDS dst
    gint_t g = (gint_t)(int*)(gv + 3 * base);
    lint_t l = (lint_t)(int*)sbuf;
    for (int k = tid; k < ndw; k += 256) {
        int slot = ((k / 3) << 2) + (k % 3);
        __builtin_amdgcn_global_load_async_to_lds_b32(g + k, l + slot, 0, 0);
    }
#else
    float* s = (float*)sbuf;
    for (int k = tid; k < ndw; k += 256) {
        int slot = ((k / 3) << 2) + (k % 3);
        s[slot] = gv[3 * base + k];
    }
#endif
    // zero-pad xyz of the 16-column tail so padded dot products stay finite
    float* sp = (float*)sbuf;
    for (int k = ndw + tid; k < 3 * cnt16; k += 256) {
        int slot = ((k / 3) << 2) + (k % 3);
        sp[slot] = 0.0f;
    }
}

__device__ __forceinline__ void wait_stage() {
#if ASYNC_STAGE
#if __has_builtin(__builtin_amdgcn_s_wait_asynccnt)
    __builtin_amdgcn_s_wait_asynccnt(0);
#else
    asm volatile("s_wait_asynccnt 0x0" ::: "memory");
#endif
#endif
}

// ---------------------------------------------------------------------------
// one chunk of 16 WMMA tiles; FULL = branch-free fast path (cnt == 256)
// ---------------------------------------------------------------------------
template <bool FULL>
__device__ __forceinline__ void do_tiles(const float4* __restrict__ cur, int base,
                                         int cnt, int cnt16, int lm, int hi,
                                         v2f afrag, float (&rmin)[8], int (&ridx)[8],
                                         unsigned* cmin) {
    float colmArr[16];
#pragma unroll
    for (int tt = 0; tt < 16; ++tt) {
        if (FULL || tt * 16 < cnt16) {
            const int col = tt * 16 + lm;
            float4 gq = cur[col];                 // one ds_load_b128
            v2f bfrag;                            // B' 4x16: K=3 component is 1
            bfrag.x = hi ? gq.z : gq.x;
            bfrag.y = hi ? 1.0f : gq.y;
            v8f cacc = {};
            cacc = __builtin_amdgcn_wmma_f32_16x16x4_f32(
                       false, afrag, false, bfrag, (short)0, cacc, false, false);

            float b2 = gq.x*gq.x + gq.y*gq.y + gq.z*gq.z;
            if (!FULL) b2 = (col < cnt) ? b2 : __builtin_inff();  // kill padded cols
            const int n = base + col;
            float colm = __builtin_inff();
#pragma unroll
            for (int r = 0; r < 8; ++r) {
                float d2 = fmaxf(cacc[r] + b2, 0.0f);
                bool tk = d2 < rmin[r];           // strict: first occurrence wins
                rmin[r] = tk ? d2 : rmin[r];
                ridx[r] = tk ? n  : ridx[r];
                colm = fminf(colm, d2);
            }
            colmArr[tt] = colm;
        }
    }
    // flush per-chunk column mins (out of the WMMA loop)
#pragma unroll
    for (int tt = 0; tt < 16; ++tt) {
        if (FULL || tt * 16 < cnt16) {
            float cm = fminf(colmArr[tt], __shfl_xor(colmArr[tt], 16, 32));
            if (hi == 0) atomicMin(&cmin[tt * 16 + lm], __float_as_uint(cm));
        }
    }
}

// ---------------------------------------------------------------------------
// WMMA cdist.  A' = (-2x,-2y,-2z,|a|^2), B' = (gx,gy,gz,1)  =>
// WMMA yields -2*a.b + |a|^2 directly; epilogue is d2 = max(c + |b|^2, 0).
// Double-buffered async LDS staging; row min+argmin via shuffle butterfly
// then packed-u64 atomicMin; column mins via LDS then global u32 atomicMin.
// ---------------------------------------------------------------------------
__global__ void k_cdist(const float* __restrict__ pv, const float* __restrict__ gv,
                        int np, int ng,
                        unsigned long long* __restrict__ rowpk,
                        unsigned* __restrict__ colmin) {
    __shared__ float4   sbuf[2][256];
    __shared__ unsigned cmin[256];

    const int tid  = threadIdx.x;
    const int lane = tid & 31;
    const int wave = tid >> 5;
    const int lm   = lane & 15;
    const int hi   = lane >> 4;
    const int m0   = blockIdx.x * 128 + wave * 16;

    int m = m0 + lm; if (m >= np) m = np - 1;      // clamp keeps EXEC full for WMMA
    float px = pv[3*m], py = pv[3*m+1], pz = pv[3*m+2];
    float a2 = px*px + py*py + pz*pz;
    v2f afrag;                                      // A' 16x4: lanes<16 K=0,1; lanes>=16 K=2,3
    afrag.x = hi ? (-2.0f * pz) : (-2.0f * px);
    afrag.y = hi ? a2           : (-2.0f * py);

    float rmin[8]; int ridx[8];
#pragma unroll
    for (int r = 0; r < 8; ++r) { rmin[r] = __builtin_inff(); ridx[r] = 0; }

    // column split across gridDim.y for occupancy
    const int segsz = (ng + (int)gridDim.y - 1) / (int)gridDim.y;
    const int y0    = (int)blockIdx.y * segsz;
    const int yend  = min(ng, y0 + segsz);
    const int nchunks = (yend > y0) ? ((yend - y0 + 255) >> 8) : 0;

    if (nchunks > 0) {
        int c0 = min(256, yend - y0);
        stage_chunk(gv, y0, c0, (c0 + 15) & ~15, &sbuf[0][0], tid);
    }

    for (int c = 0; c < nchunks; ++c) {
        const int base  = y0 + (c << 8);
        const int cnt   = min(256, yend - base);
        const int cnt16 = (cnt + 15) & ~15;
        const float4* cur = &sbuf[c & 1][0];

        cmin[tid] = INF_BITS;
        wait_stage();            // chunk c's async copies complete (per wave)
        __syncthreads();         // visible to all waves

        if (c + 1 < nchunks) {   // kick off next chunk while we compute
            const int nb = base + 256;
            const int nc = min(256, yend - nb);
            stage_chunk(gv, nb, nc, (nc + 15) & ~15, &sbuf[(c + 1) & 1][0], tid);
        }

        if (cnt == 256) do_tiles<true >(cur, base, cnt, cnt16, lm, hi, afrag, rmin, ridx, cmin);
        else            do_tiles<false>(cur, base, cnt, cnt16, lm, hi, afrag, rmin, ridx, cmin);

        __syncthreads();
        if (tid < cnt) atomicMin(&colmin[base + tid], cmin[tid]);
        __syncthreads();
    }

    // merge row min/argmin across the 16 lanes of each half (ties -> lowest idx)
#pragma unroll
    for (int s = 1; s < 16; s <<= 1) {
#pragma unroll
        for (int r = 0; r < 8; ++r) {
            float om = __shfl_xor(rmin[r], s, 32);
            int   oi = __shfl_xor(ridx[r], s, 32);
            bool tk = (om < rmin[r]) || (om == rmin[r] && oi < ridx[r]);
            rmin[r] = tk ? om : rmin[r];
            ridx[r] = tk ? oi : ridx[r];
        }
    }
    if (lm == 0) {
#pragma unroll
        for (int r = 0; r < 8; ++r) {
            int row = m0 + 8*hi + r;
            if (row < np) {
                unsigned long long pk =
                    ((unsigned long long)__float_as_uint(rmin[r]) << 32) |
                    (unsigned long long)(unsigned)ridx[r];
                atomicMin(&rowpk[row], pk);        // d2>=0 -> float order == int order
            }
        }
    }
}

// ---------------------------------------------------------------------------
// per-element loss terms -> 5 accumulators
// ---------------------------------------------------------------------------
__global__ void k_reduce(const float* __restrict__ rel, const float* __restrict__ targ,
                         const float* __restrict__ pc,
                         const unsigned long long* __restrict__ rowpk,
                         const float* __restrict__ pr_vn, const float* __restrict__ gt_vn,
                         const float* __restrict__ nsum, const float* __restrict__ deg,
                         const float* __restrict__ pv,
                         const unsigned* __restrict__ colmin,
                         float* __restrict__ acc, int np, int ng) {
    int i = blockIdx.x * blockDim.x + threadIdx.x;
    if (i < np) {
        // spatial (focal-style), gamma = 2
        float p   = fminf(fmaxf(rel[i], EPSF), 1.0f - EPSF);
        bool  pos = targ[i] > 0.5f;
        float tot = (float)np;
        float alpha = (tot - *pc) / (tot + EPSF);
        float om = 1.0f - p;
        float l  = pos ? (-alpha * om * om * logf(p))
                       : (-(1.0f - alpha) * p * p * logf(om));
        atomicAdd(&acc[0], l);

        unsigned long long pk = rowpk[i];
        float dmin = __uint_as_float((unsigned)(pk >> 32));
        int   nn   = (int)(unsigned)(pk & 0xFFFFFFFFull);
        atomicAdd(&acc[1], dmin);

        // normal consistency vs nearest gt vertex
        float pnx = pr_vn[3*i], pny = pr_vn[3*i+1], pnz = pr_vn[3*i+2];
        float pn  = fmaxf(sqrtf(pnx*pnx + pny*pny + pnz*pnz), EPSF);
        pnx /= pn; pny /= pn; pnz /= pn;
        float gnx = gt_vn[3*nn], gny = gt_vn[3*nn+1], gnz = gt_vn[3*nn+2];
        float gn  = fmaxf(sqrtf(gnx*gnx + gny*gny + gnz*gnz), EPSF);
        gnx /= gn; gny /= gn; gnz /= gn;
        float dx = pnx - gnx, dy = pny - gny, dz = pnz - gnz;
        atomicAdd(&acc[2], dx*dx + dy*dy + dz*dz);

        // uniform laplacian
        float dg = fmaxf(deg[i], 1.0f);
        float lx = nsum[3*i]   / dg - pv[3*i];
        float ly = nsum[3*i+1] / dg - pv[3*i+1];
        float lz = nsum[3*i+2] / dg - pv[3*i+2];
        atomicAdd(&acc[3], sqrtf(lx*lx + ly*ly + lz*lz));
    }
    if (i < ng) atomicAdd(&acc[4], __uint_as_float(colmin[i]));
}

__global__ void k_final(const float* __restrict__ acc, float* __restrict__ out,
                        int np, int ng) {
    if (threadIdx.x != 0 || blockIdx.x != 0) return;
    float npf = (float)np, ngf = (float)ng;
    float spatial   = acc[0] / (npf + EPSF);
    float distance  = acc[1] / npf + acc[4] / ngf;
    float normal    = acc[2] / (3.0f * npf);
    float laplacian = acc[3] / npf;
    out[0] = spatial + 1.0f * distance + 0.01f * normal + 0.1f * laplacian;
}

// ---------------------------------------------------------------------------
extern "C" void kernel_launch(void* const* d_in, const int* in_sizes, int n_in,
                              void* d_out, int out_size, void* d_ws, size_t ws_size,
                              hipStream_t stream) {
    const float* pv    = (const float*)d_in[0];   // [Np,3]
    const float* rel   = (const float*)d_in[1];   // [Np]
    const float* gv    = (const float*)d_in[2];   // [Ng,3]
    const int*   pf    = (const int*)d_in[3];     // [Fp,3]
    const int*   gf    = (const int*)d_in[4];     // [Fg,3]
    const int*   label = (const int*)d_in[5];     // [1,1,D,H,W]

    const int np  = in_sizes[0] / 3;
    const int ng  = in_sizes[2] / 3;
    const int nfp = in_sizes[3] / 3;
    const int nfg = in_sizes[4] / 3;
    int side = 1;
    while (side * side * side < in_sizes[5]) ++side;   // 96 for the given shape
    const int D = side, H = side, W = side;

    // ----- workspace layout (float words; rowpk offset is even -> 8B aligned)
    float* ws = (float*)d_ws;
    size_t o = 0;
    float* gt_vn = ws + o;  o += (size_t)3 * ng;
    float* pr_vn = ws + o;  o += (size_t)3 * np;
    float* nsum  = ws + o;  o += (size_t)3 * np;
    float* deg   = ws + o;  o += (size_t)np;
    float* targ  = ws + o;  o += (size_t)np;
    o = (o + 1) & ~(size_t)1;                        // 8-byte align
    int rp_off = (int)o;
    unsigned long long* rowpk = (unsigned long long*)(ws + o); o += (size_t)2 * np;
    int col_off = (int)o;
    unsigned* colmin = (unsigned*)(ws + o); o += (size_t)ng;
    float* pc  = ws + o; o += 1;
    float* acc = ws + o; o += 8;
    const int nwords = (int)o;

    const int BS = 256;
    // 1. init workspace
    k_init<<<(nwords + BS - 1) / BS, BS, 0, stream>>>(
        (unsigned*)d_ws, nwords, rp_off, 2 * np, col_off, ng);
    // 2. gt vertex-normal scatter
    k_face_normals<<<(nfg + BS - 1) / BS, BS, 0, stream>>>(gf, gv, gt_vn, nfg);
    // 3. pred vertex-normal scatter
    k_face_normals<<<(nfp + BS - 1) / BS, BS, 0, stream>>>(pf, pv, pr_vn, nfp);
    // 4. laplacian scatter (pred mesh)
    k_lap_scatter<<<(nfp + BS - 1) / BS, BS, 0, stream>>>(pf, pv, nsum, deg, nfp);
    // 5. label sampling + positive count
    k_targets<<<(np + BS - 1) / BS, BS, 0, stream>>>(pv, label, targ, pc, np, D, H, W);
    // 6. WMMA cdist (row strips x 8 column splits)
    dim3 cgrid((np + 127) / 128, 8);
    k_cdist<<<cgrid, BS, 0, stream>>>(pv, gv, np, ng, rowpk, colmin);
    // 7. loss-term reductions
    int nmax = np > ng ? np : ng;
    k_reduce<<<(nmax + BS - 1) / BS, BS, 0, stream>>>(
        rel, targ, pc, rowpk, pr_vn, gt_vn, nsum, deg, pv, colmin, acc, np, ng);
    // 8. combine
    k_final<<<1, 32, 0, stream>>>(acc, (float*)d_out, np, ng);
}